// PersonalizedHeteroGNN_8658654069109
// MI455X (gfx1250) — compile-verified
//
#include <hip/hip_runtime.h>
#include <hip/hip_bf16.h>

typedef __attribute__((ext_vector_type(2))) float v2f;
typedef __attribute__((ext_vector_type(8))) float v8f;

// ---- problem constants (match reference) ----
#define PN 200000
#define UN 100000
#define BN 5000
#define CN 2000
#define SN 1000
#define NN (PN + UN + BN + CN + SN)   // 308000
#define OFF_U (PN)                    // 200000
#define OFF_B (PN + UN)               // 300000
#define OFF_C (PN + UN + BN)          // 305000
#define OFF_S (PN + UN + BN + CN)     // 307000
#define E_PB 200000
#define E_PC 200000
#define E_PS 400000
#define E_UP 1000000
#define KDIM 384

// =====================================================================
// Projection GEMM: x0[0:P,0:64] = relu(product_x[P,384] @ W[384,64] + b)
// One wave per PAIR of 16-row tiles (B reused across both), 4 col-tiles,
// fp32 WMMA 16x16x4. Weights staged in LDS k-pair-interleaved so each
// B operand is a single aligned ds_load_b64 (no VGPR repacking).
// =====================================================================
__global__ __launch_bounds__(128) void hgnn_proj_kernel(
    const float* __restrict__ X, const float* __restrict__ W,
    const float* __restrict__ bias, float* __restrict__ Y, int npairs)
{
    constexpr int KC = 96;
    __shared__ v2f sWp[(KC / 2) * 64];      // 24 KB: sWp[p*64+c] = {W[2p][c], W[2p+1][c]}
    const int lane = threadIdx.x;           // 0..31
    const int wave = threadIdx.y;           // 0..3
    const int tid  = wave * 32 + lane;      // 0..127
    const int pair = blockIdx.x * 4 + wave;
    const int lo = lane & 15, hi = lane >> 4;
    const int row0 = (pair * 2) * 16 + lo;
    const int row1 = row0 + 16;

    v8f acc[2][4];
#pragma unroll
    for (int mt = 0; mt < 2; ++mt)
#pragma unroll
        for (int ct = 0; ct < 4; ++ct)
#pragma unroll
            for (int j = 0; j < 8; ++j) acc[mt][ct][j] = 0.0f;

    for (int kc = 0; kc < KDIM; kc += KC) {
        __syncthreads();
        // stage W rows [kc, kc+KC) into LDS, k-pair interleaved
        const float* Wg = W + kc * 64;
        for (int i = tid; i < (KC / 2) * 64; i += 128) {
            const int p = i >> 6, c = i & 63;
            v2f w;
            w.x = Wg[(2 * p)     * 64 + c];
            w.y = Wg[(2 * p + 1) * 64 + c];
            sWp[i] = w;
        }
        __syncthreads();
        if (pair < npairs) {
            for (int k = 0; k < KC; k += 4) {
                const int pp = (k >> 1) + hi;                    // k-pair index
                const int ko = kc + k + 2 * hi;
                // A 16x4 fp32: lanes 0-15 M=0..15 K={0,1}; lanes 16-31 K={2,3}
                const v2f a0 = *(const v2f*)(X + row0 * KDIM + ko);
                const v2f a1 = *(const v2f*)(X + row1 * KDIM + ko);
#pragma unroll
                for (int ct = 0; ct < 4; ++ct) {
                    const v2f b = sWp[pp * 64 + ct * 16 + lo];   // one ds_load_b64
                    acc[0][ct] = __builtin_amdgcn_wmma_f32_16x16x4_f32(
                        false, a0, false, b, (short)0, acc[0][ct], false, false);
                    acc[1][ct] = __builtin_amdgcn_wmma_f32_16x16x4_f32(
                        false, a1, false, b, (short)0, acc[1][ct], false, false);
                }
            }
        }
    }
    if (pair < npairs) {
#pragma unroll
        for (int mt = 0; mt < 2; ++mt) {
            const int rbase = (pair * 2 + mt) * 16;
#pragma unroll
            for (int ct = 0; ct < 4; ++ct) {
                const int col = ct * 16 + lo;
                const float bb = bias[col];
#pragma unroll
                for (int v = 0; v < 8; ++v) {
                    float val = acc[mt][ct][v] + bb;             // C/D: row = v + 8*hi
                    val = fmaxf(val, 0.0f);
                    Y[(rbase + v + 8 * hi) * 64 + col] = val;
                }
            }
        }
    }
}

// =====================================================================
// Copy user/brand/cat/shop embeddings into x0 rows [P, N) (float4)
// =====================================================================
__global__ void hgnn_copy_emb_kernel(
    const float4* __restrict__ ue, const float4* __restrict__ be,
    const float4* __restrict__ ce, const float4* __restrict__ se,
    float4* __restrict__ dst4, int total4)
{
    const int i = blockIdx.x * blockDim.x + threadIdx.x;
    if (i >= total4) return;
    const int U4 = UN * 16, B4 = BN * 16, C4 = CN * 16;
    float4 v;
    if (i < U4)                 v = ue[i];
    else if (i < U4 + B4)       v = be[i - U4];
    else if (i < U4 + B4 + C4)  v = ce[i - U4 - B4];
    else                        v = se[i - U4 - B4 - C4];
    dst4[i] = v;
}

// =====================================================================
// Zero fill (float4)
// =====================================================================
__global__ void hgnn_zero_kernel(float4* __restrict__ p, int n4)
{
    const int i = blockIdx.x * blockDim.x + threadIdx.x;
    if (i < n4) p[i] = make_float4(0.f, 0.f, 0.f, 0.f);
}

// =====================================================================
// Degree accumulation: both directions of one undirected relation
// =====================================================================
__global__ void hgnn_degree_kernel(
    const int* __restrict__ src, const int* __restrict__ dst,
    int offS, int offD, int E, float* __restrict__ deg)
{
    const int i = blockIdx.x * blockDim.x + threadIdx.x;
    if (i >= E) return;
    unsafeAtomicAdd(&deg[offS + src[i]], 1.0f);
    unsafeAtomicAdd(&deg[offD + dst[i]], 1.0f);
}

// =====================================================================
// Feature scatter-add for one undirected relation (both directions).
// blockDim = (64, 4): x = channel, y = edge-in-block. The 78.8 MB
// accumulator + 78.8 MB feature table are L2-resident (192 MB L2), so
// gathers and global_atomic_add_f32 stay on-chip.
// =====================================================================
__global__ __launch_bounds__(256) void hgnn_scatter_kernel(
    const float* __restrict__ Xin, float* __restrict__ Aggr,
    const int* __restrict__ src, const int* __restrict__ dst,
    int offS, int offD, int E)
{
    const int e = blockIdx.x * 4 + threadIdx.y;
    if (e >= E) return;
    const int c = threadIdx.x;                   // 0..63
    const int s = offS + src[e];
    const int d = offD + dst[e];
    const float vs = Xin[s * 64 + c];
    const float vd = Xin[d * 64 + c];
    unsafeAtomicAdd(&Aggr[d * 64 + c], vs);
    unsafeAtomicAdd(&Aggr[s * 64 + c], vd);
}

// =====================================================================
// Fused SAGE linear: Out = [relu]( (Aggr/max(deg,1)) @ Wl + Xin @ Wr + b )
// One wave per PAIR of 16-row tiles; fp32 WMMA; both matmuls share
// accumulators; B operands (Wl/Wr) LDS k-pair-interleaved, reused by
// both M-tiles. OD=64 (layer1, relu, in-place) or OD=32 (layer2).
// =====================================================================
template <int OD, bool RELU>
__global__ __launch_bounds__(128) void hgnn_sage_kernel(
    const float* __restrict__ Xin, const float* __restrict__ Aggr,
    const float* __restrict__ Deg,
    const float* __restrict__ Wl, const float* __restrict__ bl,
    const float* __restrict__ Wr,
    float* __restrict__ Out, int npairs)
{
    __shared__ v2f sWlp[32 * OD];                // sWlp[p*OD+c] = {Wl[2p][c], Wl[2p+1][c]}
    __shared__ v2f sWrp[32 * OD];
    __shared__ float sB[OD];
    const int lane = threadIdx.x;
    const int wave = threadIdx.y;
    const int tid  = wave * 32 + lane;
    for (int i = tid; i < 32 * OD; i += 128) {
        const int p = i / OD, c = i % OD;
        v2f w;
        w.x = Wl[(2 * p)     * OD + c];
        w.y = Wl[(2 * p + 1) * OD + c];
        sWlp[i] = w;
        v2f u;
        u.x = Wr[(2 * p)     * OD + c];
        u.y = Wr[(2 * p + 1) * OD + c];
        sWrp[i] = u;
    }
    if (tid < OD) sB[tid] = bl[tid];
    __syncthreads();

    const int pair = blockIdx.x * 4 + wave;
    if (pair >= npairs) return;                  // wave-uniform exit

    const int lo = lane & 15, hi = lane >> 4;
    const int row0 = (pair * 2) * 16 + lo;
    const int row1 = row0 + 16;
    const float rd0 = 1.0f / fmaxf(Deg[row0], 1.0f);
    const float rd1 = 1.0f / fmaxf(Deg[row1], 1.0f);

    constexpr int NCT = OD / 16;
    v8f acc[2][NCT];
#pragma unroll
    for (int mt = 0; mt < 2; ++mt)
#pragma unroll
        for (int ct = 0; ct < NCT; ++ct)
#pragma unroll
            for (int j = 0; j < 8; ++j) acc[mt][ct][j] = 0.0f;

#pragma unroll
    for (int k = 0; k < 64; k += 4) {
        const int pp = (k >> 1) + hi;
        const int ko = k + 2 * hi;
        v2f a1_0 = *(const v2f*)(Aggr + row0 * 64 + ko);
        a1_0.x *= rd0; a1_0.y *= rd0;            // mean aggregation
        const v2f a2_0 = *(const v2f*)(Xin + row0 * 64 + ko);
        v2f a1_1 = *(const v2f*)(Aggr + row1 * 64 + ko);
        a1_1.x *= rd1; a1_1.y *= rd1;
        const v2f a2_1 = *(const v2f*)(Xin + row1 * 64 + ko);
#pragma unroll
        for (int ct = 0; ct < NCT; ++ct) {
            const int col = ct * 16 + lo;
            const v2f b1 = sWlp[pp * OD + col];  // one ds_load_b64 each
            const v2f b2 = sWrp[pp * OD + col];
            acc[0][ct] = __builtin_amdgcn_wmma_f32_16x16x4_f32(
                false, a1_0, false, b1, (short)0, acc[0][ct], false, false);
            acc[0][ct] = __builtin_amdgcn_wmma_f32_16x16x4_f32(
                false, a2_0, false, b2, (short)0, acc[0][ct], false, false);
            acc[1][ct] = __builtin_amdgcn_wmma_f32_16x16x4_f32(
                false, a1_1, false, b1, (short)0, acc[1][ct], false, false);
            acc[1][ct] = __builtin_amdgcn_wmma_f32_16x16x4_f32(
                false, a2_1, false, b2, (short)0, acc[1][ct], false, false);
        }
    }

#pragma unroll
    for (int mt = 0; mt < 2; ++mt) {
        const int rbase = (pair * 2 + mt) * 16;
#pragma unroll
        for (int ct = 0; ct < NCT; ++ct) {
            const int col = ct * 16 + lo;
            const float bb = sB[col];
#pragma unroll
            for (int v = 0; v < 8; ++v) {
                float val = acc[mt][ct][v] + bb;
                if (RELU) val = fmaxf(val, 0.0f);
                Out[(rbase + v + 8 * hi) * OD + col] = val;
            }
        }
    }
}

// =====================================================================
// Host launch
// =====================================================================
extern "C" void kernel_launch(void* const* d_in, const int* in_sizes, int n_in,
                              void* d_out, int out_size, void* d_ws, size_t ws_size,
                              hipStream_t stream)
{
    const float* product_x = (const float*)d_in[0];
    const float* user_emb  = (const float*)d_in[1];
    const float* brand_emb = (const float*)d_in[2];
    const float* cat_emb   = (const float*)d_in[3];
    const float* shop_emb  = (const float*)d_in[4];
    const float* proj_W    = (const float*)d_in[5];
    const float* proj_b    = (const float*)d_in[6];
    const float* c1_Wl     = (const float*)d_in[7];
    const float* c1_bl     = (const float*)d_in[8];
    const float* c1_Wr     = (const float*)d_in[9];
    const float* c2_Wl     = (const float*)d_in[10];
    const float* c2_bl     = (const float*)d_in[11];
    const float* c2_Wr     = (const float*)d_in[12];
    const int* pb_src = (const int*)d_in[13];
    const int* pb_dst = (const int*)d_in[14];
    const int* pc_src = (const int*)d_in[15];
    const int* pc_dst = (const int*)d_in[16];
    const int* ps_src = (const int*)d_in[17];
    const int* ps_dst = (const int*)d_in[18];
    const int* up_src = (const int*)d_in[19];
    const int* up_dst = (const int*)d_in[20];

    float* x0   = (float*)d_ws;                        // [N,64] (layer1 in-place)
    float* aggr = x0 + (size_t)NN * 64;                // [N,64]
    float* deg  = aggr + (size_t)NN * 64;              // [N]
    float* out  = (float*)d_out;                       // [N,32]

    const dim3 wblk(32, 4);

    // 1) projection GEMM -> x0 rows [0, P)
    {
        const int npairs = PN / 16 / 2;                // 6250
        hgnn_proj_kernel<<<(npairs + 3) / 4, wblk, 0, stream>>>(
            product_x, proj_W, proj_b, x0, npairs);
    }

    // 2) embeddings -> x0 rows [P, N)
    {
        const int total4 = (UN + BN + CN + SN) * 16;
        hgnn_copy_emb_kernel<<<(total4 + 255) / 256, 256, 0, stream>>>(
            (const float4*)user_emb, (const float4*)brand_emb,
            (const float4*)cat_emb, (const float4*)shop_emb,
            (float4*)(x0 + (size_t)PN * 64), total4);
    }

    // 3) zero aggr + deg
    hgnn_zero_kernel<<<(NN * 64 / 4 + 255) / 256, 256, 0, stream>>>((float4*)aggr, NN * 64 / 4);
    hgnn_zero_kernel<<<(NN / 4 + 255) / 256, 256, 0, stream>>>((float4*)deg, NN / 4);

    // 4) degrees (both directions of each undirected relation)
    hgnn_degree_kernel<<<(E_PB + 255) / 256, 256, 0, stream>>>(pb_src, pb_dst, 0, OFF_B, E_PB, deg);
    hgnn_degree_kernel<<<(E_PC + 255) / 256, 256, 0, stream>>>(pc_src, pc_dst, 0, OFF_C, E_PC, deg);
    hgnn_degree_kernel<<<(E_PS + 255) / 256, 256, 0, stream>>>(ps_src, ps_dst, 0, OFF_S, E_PS, deg);
    hgnn_degree_kernel<<<(E_UP + 255) / 256, 256, 0, stream>>>(up_src, up_dst, OFF_U, 0, E_UP, deg);

    const dim3 sblk(64, 4);
    // 5) layer-1 scatter of x0 into aggr
    hgnn_scatter_kernel<<<(E_PB + 3) / 4, sblk, 0, stream>>>(x0, aggr, pb_src, pb_dst, 0, OFF_B, E_PB);
    hgnn_scatter_kernel<<<(E_PC + 3) / 4, sblk, 0, stream>>>(x0, aggr, pc_src, pc_dst, 0, OFF_C, E_PC);
    hgnn_scatter_kernel<<<(E_PS + 3) / 4, sblk, 0, stream>>>(x0, aggr, ps_src, ps_dst, 0, OFF_S, E_PS);
    hgnn_scatter_kernel<<<(E_UP + 3) / 4, sblk, 0, stream>>>(x0, aggr, up_src, up_dst, OFF_U, 0, E_UP);

    // 6) SAGE layer 1 (relu), in-place x0 -> x0
    const int spairs = NN / 16 / 2;                    // 9625
    hgnn_sage_kernel<64, true><<<(spairs + 3) / 4, wblk, 0, stream>>>(
        x0, aggr, deg, c1_Wl, c1_bl, c1_Wr, x0, spairs);

    // 7) re-zero aggr; layer-2 scatter of h1 (=x0)
    hgnn_zero_kernel<<<(NN * 64 / 4 + 255) / 256, 256, 0, stream>>>((float4*)aggr, NN * 64 / 4);
    hgnn_scatter_kernel<<<(E_PB + 3) / 4, sblk, 0, stream>>>(x0, aggr, pb_src, pb_dst, 0, OFF_B, E_PB);
    hgnn_scatter_kernel<<<(E_PC + 3) / 4, sblk, 0, stream>>>(x0, aggr, pc_src, pc_dst, 0, OFF_C, E_PC);
    hgnn_scatter_kernel<<<(E_PS + 3) / 4, sblk, 0, stream>>>(x0, aggr, ps_src, ps_dst, 0, OFF_S, E_PS);
    hgnn_scatter_kernel<<<(E_UP + 3) / 4, sblk, 0, stream>>>(x0, aggr, up_src, up_dst, OFF_U, 0, E_UP);

    // 8) SAGE layer 2 (no relu) -> d_out [N,32]
    hgnn_sage_kernel<32, false><<<(spairs + 3) / 4, wblk, 0, stream>>>(
        x0, aggr, deg, c2_Wl, c2_bl, c2_Wr, out, spairs);
}